// PositionalCharacterLevelWordSparseEncoding_31868657336782
// MI455X (gfx1250) — compile-verified
//
#include <hip/hip_runtime.h>

#ifndef __has_builtin
#define __has_builtin(x) 0
#endif

// Problem constants (fixed by the reference).
#define NUM_EMB 2048
#define MAX_POS 32
#define OUT_W   (NUM_EMB + MAX_POS)   // 2080 floats per row
#define WLEN    16                    // characters per word
#define ROWS    (32 * 512)            // B*W = 16384 rows
#define RPB     4                     // rows per block
#define BLOCK   256
#define CHUNKS  (RPB * OUT_W / 4)     // 2080 float4 (16B) chunks of real data
#define FULLIT  (CHUNKS / BLOCK)      // 8 full store iterations
#define TAIL    (CHUNKS % BLOCK)      // 32 remainder chunks
// Zero region padded to a multiple of BLOCK chunks -> branch-free zeroing.
#define ZCHUNKS (((CHUNKS + BLOCK - 1) / BLOCK) * BLOCK)   // 2304

// Async LDS->global builtin operand types (probe-confirmed signature:
// (v4i addrspace(1)*, v4i addrspace(3)*, imm offset, imm cpol)).
typedef int v4i __attribute__((vector_size(16)));
typedef __attribute__((address_space(1))) v4i g_v4i;   // global
typedef __attribute__((address_space(3))) v4i l_v4i;   // LDS

__global__ __launch_bounds__(BLOCK) void
pos_char_word_hist_kernel(const int* __restrict__ tok,
                          const int* __restrict__ pos,
                          float* __restrict__ out)
{
    __shared__ __align__(16) float lds[ZCHUNKS * 4];   // 36,864 B

    const int tid  = threadIdx.x;
    const int row0 = blockIdx.x * RPB;

    // ---- Phase 1: zero the per-row histograms (9 unconditional b128 DS stores) ----
    float4* lds4 = reinterpret_cast<float4*>(lds);
    #pragma unroll
    for (int i = 0; i < ZCHUNKS / BLOCK; ++i)
        lds4[tid + i * BLOCK] = float4{0.f, 0.f, 0.f, 0.f};
    __syncthreads();

    // ---- Phase 2: scatter increments (ds_add_f32 LDS atomics) ----
    // 4 rows x (16 token + 16 position) = 128 scatter ops, one per lane.
    if (tid < RPB * 2 * WLEN) {
        const int r   = tid >> 5;          // row within block
        const int e   = tid & 31;          // element within row
        const int row = row0 + r;
        if (e < WLEN) {
            const int id = tok[row * WLEN + e];
            if (id != 0)                   // padding_idx contributes zero
                atomicAdd(&lds[r * OUT_W + id], 1.0f);
        } else {
            const int id = pos[row * WLEN + (e - WLEN)];
            if (id != 0)
                atomicAdd(&lds[r * OUT_W + NUM_EMB + id], 1.0f);
        }
    }
    __syncthreads();

    // ---- Phase 3: stream LDS -> HBM via async b128 stores (ASYNCcnt) ----
    float* gbase = out + (size_t)row0 * OUT_W;
    #pragma unroll
    for (int k = 0; k < FULLIT; ++k) {
        const int c = tid + k * BLOCK;
#if __has_builtin(__builtin_amdgcn_global_store_async_from_lds_b128)
        __builtin_amdgcn_global_store_async_from_lds_b128(
            (g_v4i*)(gbase + c * 4), (l_v4i*)&lds[c * 4], 0, 0);
#else
        asm volatile("global_store_async_from_lds_b128 %0, %1, off"
                     :: "v"((unsigned long long)(gbase + c * 4)),
                        "v"((unsigned)(unsigned long long)
                            ( __attribute__((address_space(3))) float*)&lds[c * 4])
                     : "memory");
#endif
    }
    if (tid < TAIL) {                       // chunks 2048..2079
        const int c = FULLIT * BLOCK + tid;
#if __has_builtin(__builtin_amdgcn_global_store_async_from_lds_b128)
        __builtin_amdgcn_global_store_async_from_lds_b128(
            (g_v4i*)(gbase + c * 4), (l_v4i*)&lds[c * 4], 0, 0);
#else
        asm volatile("global_store_async_from_lds_b128 %0, %1, off"
                     :: "v"((unsigned long long)(gbase + c * 4)),
                        "v"((unsigned)(unsigned long long)
                            ( __attribute__((address_space(3))) float*)&lds[c * 4])
                     : "memory");
#endif
    }

#if __has_builtin(__builtin_amdgcn_s_wait_asynccnt)
    __builtin_amdgcn_s_wait_asynccnt(0);
#else
    asm volatile("s_wait_asynccnt 0" ::: "memory");
#endif
}

extern "C" void kernel_launch(void* const* d_in, const int* in_sizes, int n_in,
                              void* d_out, int out_size, void* d_ws, size_t ws_size,
                              hipStream_t stream) {
    (void)in_sizes; (void)n_in; (void)d_ws; (void)ws_size; (void)out_size;
    const int* tok = (const int*)d_in[0];   // token_ids    (B,W,L) int32
    const int* pos = (const int*)d_in[1];   // position_ids (B,W,L) int32
    float* out     = (float*)d_out;         // (B,W,2080) float32

    dim3 grid(ROWS / RPB);                  // 4096 blocks
    dim3 block(BLOCK);                      // 256 threads = 8 wave32
    hipLaunchKernelGGL(pos_char_word_hist_kernel, grid, block, 0, stream,
                       tok, pos, out);
}